// GINWrapper_86870008529629
// MI455X (gfx1250) — compile-verified
//
#include <hip/hip_runtime.h>
#include <hip/hip_bf16.h>

// GIN layer for MI455X (gfx1250, wave32):
//   agg = (1+eps)*x ; agg[dst] += x[src] over edges ; out = relu(agg@W1+b1)@W2+b2
//
// Kernel 1: init agg = (1+eps)*x            (stream, float4)
// Kernel 2: edge scatter-add                (global_atomic_add_f32; NT loads for
//                                            the once-touched 25.6MB edge list so
//                                            x/agg stay L2-resident)
// Kernel 3: fused 2-layer MLP, V_WMMA_F32_16X16X4_F32, M=32 per wave (two 16-row
//           subtiles share every layer-1 B load), weights staged K-pair-interleaved
//           in LDS so each B fragment is ONE aligned ds_load_b64.

typedef __attribute__((ext_vector_type(2))) float v2f;
typedef __attribute__((ext_vector_type(8))) float v8f;

#define D 64
#define MLP_WAVES 4   // 4 waves * 32 rows = 128 rows/block; LDS = 16+16+16 = 48 KB

// ---------------- Kernel 1: agg = (1+eps) * x ----------------
__global__ __launch_bounds__(256) void gin_init(const float* __restrict__ x,
                                                const float* __restrict__ eps,
                                                float* __restrict__ agg,
                                                int n4) {
  int t = blockIdx.x * 256 + threadIdx.x;
  if (t >= n4) return;
  const float s = 1.0f + eps[0];
  float4 v = reinterpret_cast<const float4*>(x)[t];
  float4 r;
  r.x = v.x * s; r.y = v.y * s; r.z = v.z * s; r.w = v.w * s;
  reinterpret_cast<float4*>(agg)[t] = r;
}

// ---------------- Kernel 2: agg[dst] += x[src] ----------------
// 16 threads cooperate on one edge (4 floats each): float4 load + 4 f32 atomics.
__global__ __launch_bounds__(256) void gin_scatter(const float* __restrict__ x,
                                                   const long long* __restrict__ ei,
                                                   float* __restrict__ agg,
                                                   int E) {
  int t = blockIdx.x * 256 + threadIdx.x;
  int e = t >> 4;
  if (e >= E) return;
  int d = (t & 15) * 4;
  long long src = __builtin_nontemporal_load(ei + e);             // streamed once
  long long dst = __builtin_nontemporal_load(ei + (size_t)E + e); // streamed once
  float4 v = *reinterpret_cast<const float4*>(x + (size_t)src * D + d);
  float* ap = agg + (size_t)dst * D + d;
  atomicAdd(ap + 0, v.x);
  atomicAdd(ap + 1, v.y);
  atomicAdd(ap + 2, v.z);
  atomicAdd(ap + 3, v.w);
}

// ---------------- Kernel 3: out = relu(agg@W1+b1)@W2+b2 ----------------
// V_WMMA_F32_16X16X4_F32 wave32 layouts:
//   A 16x4 : lanes 0-15 rows M=0..15 hold K={0,1}; lanes 16-31 hold K={2,3}
//   B 4x16 : lanes 0-15 cols N=0..15 hold K={0,1}; lanes 16-31 hold K={2,3}
//   C/D    : VGPR v: lanes 0-15 -> (M=v, N=lane), lanes 16-31 -> (M=v+8, N=lane-16)
// Weights staged interleaved: element (krow, n) at LDS index
//   (krow>>1)*128 + 2*n + (krow&1)  => per-lane {k, k+1} pair is contiguous b64.
__global__ __launch_bounds__(128) void gin_mlp(const float* __restrict__ agg,
                                               const float* __restrict__ W1,
                                               const float* __restrict__ b1,
                                               const float* __restrict__ W2,
                                               const float* __restrict__ b2,
                                               float* __restrict__ out,
                                               int npairs) {
  __shared__ float sW1[D * D];              // 16 KB, interleaved layout
  __shared__ float sW2[D * D];              // 16 KB, interleaved layout
  __shared__ float sH[MLP_WAVES][16 * D];   // 16 KB, per-wave 16x64 stage (reused)

  const int tid = threadIdx.x;
  for (int i = tid; i < D * D; i += 128) {
    const int krow = i >> 6;
    const int n    = i & 63;
    const int idx  = (krow >> 1) * 128 + n * 2 + (krow & 1);
    sW1[idx] = W1[i];
    sW2[idx] = W2[i];
  }
  __syncthreads();

  const int wave = tid >> 5;
  const int lane = tid & 31;
  const int half = lane >> 4;      // K-pair select (0 -> K{0,1}, 1 -> K{2,3})
  const int l16  = lane & 15;      // row (A) / col (B,C)

  const int pair    = blockIdx.x * MLP_WAVES + wave;     // 32-row block id
  const bool active = pair < npairs;
  const int pclamp  = active ? pair : (npairs - 1);      // keep EXEC uniform
  const int rowbase = pclamp * 32;

  // ---- Layer 1: two 16-row subtiles, B loaded once per (k, nt) ----
  v2f a0[16], a1[16];
  {
    const float* r0 = agg + (size_t)(rowbase + l16) * D + half * 2;
    const float* r1 = r0 + 16 * D;
#pragma unroll
    for (int k = 0; k < 16; ++k) {
      a0[k] = *reinterpret_cast<const v2f*>(r0 + 4 * k);   // global_load_b64
      a1[k] = *reinterpret_cast<const v2f*>(r1 + 4 * k);
    }
  }

  float* hrow = &sH[wave][0];
  v8f c1sav[4];
#pragma unroll
  for (int nt = 0; nt < 4; ++nt) {
    const int n0 = nt * 16;
    v8f c0 = {}, c1 = {};
#pragma unroll
    for (int k = 0; k < 16; ++k) {
      // B fragment: one aligned ds_load_b64 from interleaved weights
      v2f b = *reinterpret_cast<const v2f*>(&sW1[(2 * k + half) * 128 + 2 * (n0 + l16)]);
      c0 = __builtin_amdgcn_wmma_f32_16x16x4_f32(false, a0[k], false, b,
                                                 (short)0, c0, false, false);
      c1 = __builtin_amdgcn_wmma_f32_16x16x4_f32(false, a1[k], false, b,
                                                 (short)0, c1, false, false);
    }
    const float bias = b1[n0 + l16];
#pragma unroll
    for (int v = 0; v < 8; ++v) {
      float val = c0[v] + bias;
      val = val > 0.0f ? val : 0.0f;                 // ReLU
      hrow[(v + 8 * half) * D + n0 + l16] = val;     // stage subtile 0
      c1sav[nt][v] = c1[v];                          // park subtile 1
    }
  }

  // ---- Layer 2, subtile 0 ----
  asm volatile("s_wait_dscnt 0" ::: "memory");       // cross-lane LDS handoff
  v2f a2[16];
#pragma unroll
  for (int k = 0; k < 16; ++k)
    a2[k] = *reinterpret_cast<const v2f*>(&hrow[l16 * D + 4 * k + half * 2]);

#pragma unroll
  for (int nt = 0; nt < 4; ++nt) {
    const int n0 = nt * 16;
    v8f c = {};
#pragma unroll
    for (int k = 0; k < 16; ++k) {
      v2f b = *reinterpret_cast<const v2f*>(&sW2[(2 * k + half) * 128 + 2 * (n0 + l16)]);
      c = __builtin_amdgcn_wmma_f32_16x16x4_f32(false, a2[k], false, b,
                                                (short)0, c, false, false);
    }
    if (active) {   // wave-uniform; WMMA already done with full EXEC
      const float bias = b2[n0 + l16];
#pragma unroll
      for (int v = 0; v < 8; ++v)
        out[(size_t)(rowbase + v + 8 * half) * D + n0 + l16] = c[v] + bias;
    }
  }

  // ---- Stage subtile 1 (per-wave DS ops are in-order; fence vs compiler) ----
  asm volatile("s_wait_dscnt 0" ::: "memory");
#pragma unroll
  for (int nt = 0; nt < 4; ++nt) {
    const int n0 = nt * 16;
    const float bias = b1[n0 + l16];
#pragma unroll
    for (int v = 0; v < 8; ++v) {
      float val = c1sav[nt][v] + bias;
      val = val > 0.0f ? val : 0.0f;
      hrow[(v + 8 * half) * D + n0 + l16] = val;
    }
  }
  asm volatile("s_wait_dscnt 0" ::: "memory");

  // ---- Layer 2, subtile 1 ----
#pragma unroll
  for (int k = 0; k < 16; ++k)
    a2[k] = *reinterpret_cast<const v2f*>(&hrow[l16 * D + 4 * k + half * 2]);

#pragma unroll
  for (int nt = 0; nt < 4; ++nt) {
    const int n0 = nt * 16;
    v8f c = {};
#pragma unroll
    for (int k = 0; k < 16; ++k) {
      v2f b = *reinterpret_cast<const v2f*>(&sW2[(2 * k + half) * 128 + 2 * (n0 + l16)]);
      c = __builtin_amdgcn_wmma_f32_16x16x4_f32(false, a2[k], false, b,
                                                (short)0, c, false, false);
    }
    if (active) {
      const float bias = b2[n0 + l16];
#pragma unroll
      for (int v = 0; v < 8; ++v)
        out[(size_t)(rowbase + 16 + v + 8 * half) * D + n0 + l16] = c[v] + bias;
    }
  }
}

extern "C" void kernel_launch(void* const* d_in, const int* in_sizes, int n_in,
                              void* d_out, int out_size, void* d_ws, size_t ws_size,
                              hipStream_t stream) {
  const float*     x   = (const float*)d_in[0];
  const long long* ei  = (const long long*)d_in[1];   // int64 edge_index [2, E]
  const float*     W1  = (const float*)d_in[2];
  const float*     b1  = (const float*)d_in[3];
  const float*     W2  = (const float*)d_in[4];
  const float*     b2  = (const float*)d_in[5];
  const float*     eps = (const float*)d_in[6];
  float*           out = (float*)d_out;

  const int N = in_sizes[0] / D;
  const int E = in_sizes[1] / 2;

  float* agg = (float*)d_ws;   // N*D floats of scratch

  // 1) agg = (1+eps)*x
  const int n4 = N * (D / 4);
  gin_init<<<(n4 + 255) / 256, 256, 0, stream>>>(x, eps, agg, n4);

  // 2) scatter-add: 16 threads per edge
  gin_scatter<<<(E + 15) / 16, 256, 0, stream>>>(x, ei, agg, E);

  // 3) fused WMMA MLP, 32 rows per wave
  const int npairs = (N + 31) / 32;
  gin_mlp<<<(npairs + MLP_WAVES - 1) / MLP_WAVES, 128, 0, stream>>>(
      agg, W1, b1, W2, b2, out, npairs);
}